// Atomwise_20134806684373
// MI455X (gfx1250) — compile-verified
//
#include <hip/hip_runtime.h>

typedef __attribute__((ext_vector_type(2))) float v2f;
typedef __attribute__((ext_vector_type(8))) float v8f;

#define B_DIM 2048
#define A_DIM 96
#define NIN 128
#define NHID 64
#define ROWS_PER_BLOCK 128
#define THREADS 256
#define LSTR 130  /* 128 + 2 floats pad: lane stride 130 -> banks 2 apart, conflict-free */

__device__ __forceinline__ float sspf(float x) {
    // softplus(x) - log(2): fmax(x,0) + log(1 + exp(-|x|)) - log(2)
    // fast-path transcendentals: abs err ~<= 6e-8, fine after weighted 64-term sum
    return fmaxf(x, 0.0f) + __logf(1.0f + __expf(-fabsf(x))) - 0.69314718055994530942f;
}

__global__ void zero_out_kernel(float* __restrict__ out, int n) {
    int i = blockIdx.x * blockDim.x + threadIdx.x;
    if (i < n) out[i] = 0.0f;
}

__launch_bounds__(THREADS)
__global__ void atomwise_kernel(const float* __restrict__ rep,
                                const int*   __restrict__ zatom,
                                const float* __restrict__ mask,
                                const float* __restrict__ W1,
                                const float* __restrict__ b1,
                                const float* __restrict__ W2,
                                const float* __restrict__ b2,
                                const float* __restrict__ atomref,
                                const float* __restrict__ mean,
                                const float* __restrict__ stddev,
                                float* __restrict__ out) {
    __shared__ float repLds[ROWS_PER_BLOCK * LSTR];   // [localRow][k], padded
    __shared__ float w1Lds[NHID * LSTR];              // transposed: [h][k], padded

    const int tid  = threadIdx.x;
    const int wave = tid >> 5;
    const int lane = tid & 31;
    const int j    = lane & 15;        // column-in-tile / row-in-half
    const int koff = (lane >> 4) * 2;  // K sub-offset per WMMA fp32 layout
    const int row0 = blockIdx.x * ROWS_PER_BLOCK;

    // ---- Stage rep tile: 128 rows x 128 floats, coalesced float4 loads ----
    // 4096 float4 total, 16 per thread
    #pragma unroll
    for (int it = 0; it < 16; ++it) {
        int idx4 = it * THREADS + tid;     // global float4 index in tile
        int r    = idx4 >> 5;              // 32 float4 per row
        int c4   = idx4 & 31;
        float4 v = reinterpret_cast<const float4*>(rep + (size_t)(row0 + r) * NIN)[c4];
        float* dst = &repLds[r * LSTR + c4 * 4];
        dst[0] = v.x; dst[1] = v.y; dst[2] = v.z; dst[3] = v.w;
    }
    // ---- Stage W1 transposed: W1[i][h] (row-major i) -> w1Lds[h*LSTR + i] ----
    // 8192 floats, 32 per thread, coalesced global reads
    #pragma unroll
    for (int it = 0; it < 32; ++it) {
        int idx = it * THREADS + tid;      // = i*64 + h
        int i   = idx >> 6;
        int h   = idx & 63;
        w1Lds[h * LSTR + i] = W1[idx];
    }
    __syncthreads();

    // ---- 16x64 GEMM tile per wave via V_WMMA_F32_16X16X4_F32 ----
    v8f c0 = {}, c1 = {}, c2 = {}, c3 = {};
    const float* aBase  = &repLds[(wave * 16 + j) * LSTR];
    const float* bBase0 = &w1Lds[( 0 + j) * LSTR];
    const float* bBase1 = &w1Lds[(16 + j) * LSTR];
    const float* bBase2 = &w1Lds[(32 + j) * LSTR];
    const float* bBase3 = &w1Lds[(48 + j) * LSTR];

    #pragma unroll 8
    for (int k = 0; k < NIN; k += 4) {
        v2f a   = *reinterpret_cast<const v2f*>(aBase  + k + koff);
        v2f bb0 = *reinterpret_cast<const v2f*>(bBase0 + k + koff);
        v2f bb1 = *reinterpret_cast<const v2f*>(bBase1 + k + koff);
        v2f bb2 = *reinterpret_cast<const v2f*>(bBase2 + k + koff);
        v2f bb3 = *reinterpret_cast<const v2f*>(bBase3 + k + koff);
        c0 = __builtin_amdgcn_wmma_f32_16x16x4_f32(false, a, false, bb0, (short)0, c0, false, false);
        c1 = __builtin_amdgcn_wmma_f32_16x16x4_f32(false, a, false, bb1, (short)0, c1, false, false);
        c2 = __builtin_amdgcn_wmma_f32_16x16x4_f32(false, a, false, bb2, (short)0, c2, false, false);
        c3 = __builtin_amdgcn_wmma_f32_16x16x4_f32(false, a, false, bb3, (short)0, c3, false, false);
    }

    // ---- Fused epilogue: +b1, shifted softplus, dot with W2 ----
    float s[8] = {0.f, 0.f, 0.f, 0.f, 0.f, 0.f, 0.f, 0.f};
    {
        float w2v = W2[ 0 + j], b1v = b1[ 0 + j];
        #pragma unroll
        for (int r = 0; r < 8; ++r) s[r] += w2v * sspf(c0[r] + b1v);
    }
    {
        float w2v = W2[16 + j], b1v = b1[16 + j];
        #pragma unroll
        for (int r = 0; r < 8; ++r) s[r] += w2v * sspf(c1[r] + b1v);
    }
    {
        float w2v = W2[32 + j], b1v = b1[32 + j];
        #pragma unroll
        for (int r = 0; r < 8; ++r) s[r] += w2v * sspf(c2[r] + b1v);
    }
    {
        float w2v = W2[48 + j], b1v = b1[48 + j];
        #pragma unroll
        for (int r = 0; r < 8; ++r) s[r] += w2v * sspf(c3[r] + b1v);
    }

    // ---- Butterfly reduction over the 16 lanes of each half-group ----
    #pragma unroll
    for (int r = 0; r < 8; ++r) {
        #pragma unroll
        for (int off = 1; off < 16; off <<= 1)
            s[r] += __shfl_xor(s[r], off, 32);
    }

    // ---- Scale/shift, atomref, mask, segmented sum via atomics ----
    if (j < 8) {
        int localRow = wave * 16 + j + ((lane >> 4) ? 8 : 0);
        int row = row0 + localRow;                    // global atom-row index
        float yi = (s[j] + b2[0]) * stddev[0] + mean[0];
        yi += atomref[zatom[row]];                    // N_OUT == 1
        yi *= mask[row];
        atomicAdd(&out[row / A_DIM], yi);
    }
}

extern "C" void kernel_launch(void* const* d_in, const int* in_sizes, int n_in,
                              void* d_out, int out_size, void* d_ws, size_t ws_size,
                              hipStream_t stream) {
    const float* rep     = (const float*)d_in[0];
    const int*   zatom   = (const int*)  d_in[1];
    const float* mask    = (const float*)d_in[2];
    const float* W1      = (const float*)d_in[3];
    const float* b1      = (const float*)d_in[4];
    const float* W2      = (const float*)d_in[5];
    const float* b2      = (const float*)d_in[6];
    const float* atomref = (const float*)d_in[7];
    const float* mean    = (const float*)d_in[8];
    const float* stddev  = (const float*)d_in[9];
    float* out = (float*)d_out;

    zero_out_kernel<<<(out_size + 255) / 256, 256, 0, stream>>>(out, out_size);

    int blocks = (B_DIM * A_DIM) / ROWS_PER_BLOCK;  // 1536
    atomwise_kernel<<<blocks, THREADS, 0, stream>>>(rep, zatom, mask, W1, b1, W2, b2,
                                                    atomref, mean, stddev, out);
    (void)in_sizes; (void)n_in; (void)d_ws; (void)ws_size;
}